// warp_layer_3d_5162550690396
// MI455X (gfx1250) — compile-verified
//
#include <hip/hip_runtime.h>
#include <hip/hip_bf16.h>
#include <stdint.h>

#define GRID3 512          // 8*8*8
#define FLOW_PER_B 1536    // 3 * 512
#define VOX 128
#define VOXSQ (VOX * VOX)
#define VOXCUBE (VOX * VOX * VOX)

// ---------------------------------------------------------------------------
// Kernel 1: per-batch cumsum of the coarse flow + offsets.
//   x layout: [B][3][8][8][8]; channel 0 cumsum along x (axis 3),
//   channel 1 along y (axis 2), channel 2 along z (axis 1).
// One block per batch, 192 threads = one thread per (channel, line).
// ---------------------------------------------------------------------------
__global__ __launch_bounds__(192) void cumsum_kernel(
    const float* __restrict__ x,
    const float* __restrict__ ox,
    const float* __restrict__ oy,
    const float* __restrict__ oz,
    float* __restrict__ flow) {
  const int b = blockIdx.x;
  const int t = threadIdx.x;
  if (t >= 192) return;
  const int c = t / 64;        // channel
  const int line = t % 64;
  const int i = line / 8;
  const int j = line % 8;

  const float off = (c == 0) ? ox[0] : ((c == 1) ? oy[0] : oz[0]);
  const float* s = x + ((size_t)b * 3 + c) * GRID3;
  float* d = flow + ((size_t)b * 3 + c) * GRID3;

  float acc = off;
  if (c == 0) {                 // along x: line = (z=i, y=j), stride 1
    const int base = i * 64 + j * 8;
#pragma unroll
    for (int k = 0; k < 8; ++k) { acc += s[base + k]; d[base + k] = acc; }
  } else if (c == 1) {          // along y: line = (z=i, x=j), stride 8
    const int base = i * 64 + j;
#pragma unroll
    for (int k = 0; k < 8; ++k) { acc += s[base + k * 8]; d[base + k * 8] = acc; }
  } else {                      // along z: line = (y=i, x=j), stride 64
    const int base = i * 8 + j;
#pragma unroll
    for (int k = 0; k < 8; ++k) { acc += s[base + k * 64]; d[base + k * 64] = acc; }
  }
}

// ---------------------------------------------------------------------------
// Kernel 2: upsample flow (clamped half-pixel trilerp, 8^3 -> 128^3) and
// trilinear grid-sample gather from src. One block = (batch b, plane z,
// two y-rows), 256 threads (8 waves on CDNA5 wave32).
// Coarse flow (6 KB) is staged to LDS with CDNA5 async global->LDS loads;
// the streamed output is stored non-temporally so the 128 MiB source volume
// stays resident in the 192 MB L2 for the gather taps.
// ---------------------------------------------------------------------------
__global__ __launch_bounds__(256) void warp_kernel(
    const float* __restrict__ src,
    const float* __restrict__ flow,
    float* __restrict__ out) {
  __shared__ float sflow[FLOW_PER_B];

  const int b = blockIdx.z;
  const int z = blockIdx.y;
  const int t = threadIdx.x;
  const int x = t & 127;
  const int y = blockIdx.x * 2 + (t >> 7);

  // ---- async stage of this batch's coarse flow into LDS -------------------
  {
    const float* g = flow + (size_t)b * FLOW_PER_B + t;   // thread t covers t + 256*k
    // For a flat pointer into LDS, addr[31:0] IS the LDS byte offset
    // (ISA 10.2: LDS aperture lives in addr[63:32]).  Deriving the offset
    // from the real sflow pointer (ptrtoint) escapes the array so the
    // "memory" clobber keeps the later ds_loads alive; we also pass the
    // flat pointer itself as an (unused) operand for good measure.
    void* sp = (void*)sflow;
    unsigned lofs = (unsigned)(uintptr_t)sp + (unsigned)(t * 4);
    unsigned long long ga = (unsigned long long)(uintptr_t)g;
    // INST_OFFSET is added to BOTH the LDS and global addresses (ISA 8.2),
    // so one base pair serves all six 1 KiB-strided transfers.
    asm volatile(
        "global_load_async_to_lds_b32 %0, %1, off\n\t"
        "global_load_async_to_lds_b32 %0, %1, off offset:1024\n\t"
        "global_load_async_to_lds_b32 %0, %1, off offset:2048\n\t"
        "global_load_async_to_lds_b32 %0, %1, off offset:3072\n\t"
        "global_load_async_to_lds_b32 %0, %1, off offset:4096\n\t"
        "global_load_async_to_lds_b32 %0, %1, off offset:5120\n\t"
        "s_wait_asynccnt 0"
        :
        : "v"(lofs), "v"(ga), "v"(sp)
        : "memory");
  }
  __syncthreads();

  // ---- trilinear upsample coords (jax.image.resize linear, half-pixel) ----
  // u = (o + 0.5)/16 - 0.5, boundary renormalization == clamp to [0, 7].
  float ux = fminf(fmaxf((x + 0.5f) * 0.0625f - 0.5f, 0.0f), 7.0f);
  float uy = fminf(fmaxf((y + 0.5f) * 0.0625f - 0.5f, 0.0f), 7.0f);
  float uz = fminf(fmaxf((z + 0.5f) * 0.0625f - 0.5f, 0.0f), 7.0f);
  int ix = min((int)ux, 6); float fx = ux - (float)ix;
  int iy = min((int)uy, 6); float fy = uy - (float)iy;
  int iz = min((int)uz, 6); float fz = uz - (float)iz;

  const float gx0 = 1.0f - fx, gy0 = 1.0f - fy, gz0 = 1.0f - fz;
  const float w000 = gz0 * gy0 * gx0, w001 = gz0 * gy0 * fx;
  const float w010 = gz0 * fy  * gx0, w011 = gz0 * fy  * fx;
  const float w100 = fz  * gy0 * gx0, w101 = fz  * gy0 * fx;
  const float w110 = fz  * fy  * gx0, w111 = fz  * fy  * fx;

  const int base = iz * 64 + iy * 8 + ix;
  float fvec[3];
#pragma unroll
  for (int c = 0; c < 3; ++c) {
    const float* s = sflow + c * GRID3 + base;
    fvec[c] = w000 * s[0]  + w001 * s[1]  +
              w010 * s[8]  + w011 * s[9]  +
              w100 * s[64] + w101 * s[65] +
              w110 * s[72] + w111 * s[73];
  }

  // ---- grid_sample_3d: g = ((f + 1)*128 - 1)*0.5 = 64*f + 63.5 ------------
  const float gx = fvec[0] * 64.0f + 63.5f;
  const float gy = fvec[1] * 64.0f + 63.5f;
  const float gz = fvec[2] * 64.0f + 63.5f;

  const float xf = floorf(gx); const int x0 = (int)xf; const float tx = gx - xf;
  const float yf = floorf(gy); const int y0 = (int)yf; const float ty = gy - yf;
  const float zf = floorf(gz); const int z0 = (int)zf; const float tz = gz - zf;

  // Per-axis masked weights (zero padding) + clamped indices, as in reference.
  const float wx0 = (x0 >= 0 && x0 < VOX)         ? (1.0f - tx) : 0.0f;
  const float wx1 = (x0 + 1 >= 0 && x0 + 1 < VOX) ? tx          : 0.0f;
  const float wy0 = (y0 >= 0 && y0 < VOX)         ? (1.0f - ty) : 0.0f;
  const float wy1 = (y0 + 1 >= 0 && y0 + 1 < VOX) ? ty          : 0.0f;
  const float wz0 = (z0 >= 0 && z0 < VOX)         ? (1.0f - tz) : 0.0f;
  const float wz1 = (z0 + 1 >= 0 && z0 + 1 < VOX) ? tz          : 0.0f;

  const int xc0 = min(max(x0, 0), VOX - 1), xc1 = min(max(x0 + 1, 0), VOX - 1);
  const int yc0 = min(max(y0, 0), VOX - 1), yc1 = min(max(y0 + 1, 0), VOX - 1);
  const int zc0 = min(max(z0, 0), VOX - 1), zc1 = min(max(z0 + 1, 0), VOX - 1);

  const float* vb = src + (size_t)b * VOXCUBE;
  const float* r00 = vb + (size_t)(zc0 * VOX + yc0) * VOX;
  const float* r01 = vb + (size_t)(zc0 * VOX + yc1) * VOX;
  const float* r10 = vb + (size_t)(zc1 * VOX + yc0) * VOX;
  const float* r11 = vb + (size_t)(zc1 * VOX + yc1) * VOX;

  float acc;
  acc  = wz0 * wy0 * (wx0 * r00[xc0] + wx1 * r00[xc1]);
  acc += wz0 * wy1 * (wx0 * r01[xc0] + wx1 * r01[xc1]);
  acc += wz1 * wy0 * (wx0 * r10[xc0] + wx1 * r10[xc1]);
  acc += wz1 * wy1 * (wx0 * r11[xc0] + wx1 * r11[xc1]);

  // Streamed, never-reread output: non-temporal store (TH=NT) keeps the
  // gather working set (src volume) resident in L2.
  __builtin_nontemporal_store(acc, &out[(((size_t)b * VOX + z) * VOX + y) * VOX + x]);
}

// ---------------------------------------------------------------------------
extern "C" void kernel_launch(void* const* d_in, const int* in_sizes, int n_in,
                              void* d_out, int out_size, void* d_ws, size_t ws_size,
                              hipStream_t stream) {
  const float* x    = (const float*)d_in[0];   // (B,3,8,8,8)
  const float* srcb = (const float*)d_in[1];   // (B,128,128,128)
  const float* ox   = (const float*)d_in[2];   // scalar
  const float* oy   = (const float*)d_in[3];   // scalar
  const float* oz   = (const float*)d_in[4];   // scalar
  float* out  = (float*)d_out;
  float* flow = (float*)d_ws;                  // (B,3,8,8,8) cumsum result

  const int B = in_sizes[0] / FLOW_PER_B;      // = 16

  hipLaunchKernelGGL(cumsum_kernel, dim3(B), dim3(192), 0, stream,
                     x, ox, oy, oz, flow);
  hipLaunchKernelGGL(warp_kernel, dim3(VOX / 2, VOX, B), dim3(256), 0, stream,
                     srcb, flow, out);
}